// GatherTopKIndicesOnIndexed_19430432047292
// MI455X (gfx1250) — compile-verified
//
#include <hip/hip_runtime.h>
#include <hip/hip_bf16.h>
#include <stdint.h>

#ifndef __has_builtin
#define __has_builtin(x) 0
#endif

namespace {
constexpr int KTOP = 1000;   // top-k kept per batch
constexpr int BATCH = 64;    // segments
constexpr int NPB = 16384;   // candidates per segment
constexpr int NCLS = 80;     // classes
constexpr int SORT_THREADS = 1024;
constexpr int EPT = NPB / SORT_THREADS;          // 16 elements per thread
constexpr int ROWS_PER_BLOCK = 64;               // gather rows staged per block
constexpr int CHUNKS_SCORES = ROWS_PER_BLOCK * (NCLS / 4);  // 1280 float4 chunks
constexpr int CHUNKS_TOTAL = CHUNKS_SCORES + ROWS_PER_BLOCK; // + 64 box chunks
}

// Builtin parameter types (from hipcc diagnostic): pointers to GCC-style
// int __vector(4) in address space 1 (global) / 3 (LDS).
typedef int v4i __attribute__((vector_size(16)));
typedef __attribute__((address_space(1))) v4i gv4i;  // global int4
typedef __attribute__((address_space(3))) v4i lv4i;  // LDS int4
typedef __attribute__((address_space(3))) void lvoid;

// ---- CDNA5 async global->LDS / LDS->global DMA helpers (ASYNCcnt) ----------
__device__ __forceinline__ void async_copy_g2l_b128(const void* g, void* l) {
#if __has_builtin(__builtin_amdgcn_global_load_async_to_lds_b128)
  __builtin_amdgcn_global_load_async_to_lds_b128(
      (gv4i*)(uintptr_t)g, (lv4i*)l, /*offset=*/0, /*cpol=*/0);
#else
  unsigned loff = (unsigned)(size_t)(lvoid*)l;  // 32-bit LDS byte offset
  asm volatile("global_load_async_to_lds_b128 %0, %1, off"
               :: "v"(loff), "v"(g) : "memory");
#endif
}

__device__ __forceinline__ void async_copy_l2g_b128(void* g, const void* l) {
#if __has_builtin(__builtin_amdgcn_global_store_async_from_lds_b128)
  __builtin_amdgcn_global_store_async_from_lds_b128(
      (gv4i*)(uintptr_t)g, (lv4i*)(void*)l, /*offset=*/0, /*cpol=*/0);
#else
  unsigned loff = (unsigned)(size_t)(lvoid*)(void*)l;
  asm volatile("global_store_async_from_lds_b128 %1, %0, off"
               :: "v"(loff), "v"(g) : "memory");
#endif
}

__device__ __forceinline__ void wait_async_zero() {
#if __has_builtin(__builtin_amdgcn_s_wait_asynccnt)
  __builtin_amdgcn_s_wait_asynccnt(0);
#else
  asm volatile("s_wait_asynccnt 0" ::: "memory");
#endif
}

// ---------------------------------------------------------------------------
// Kernel 1: one workgroup per batch segment. Pack key = (fg_bits<<32)|(~idx)
// (fg >= 0 so IEEE bits order == value order; ~idx makes ties resolve to the
// LOWER index first when sorting DESCENDING — matches jax.lax.top_k).
// Full 16384-element bitonic sort in 128 KB of LDS (needs CDNA5's 320 KB WGP
// LDS), then emit the global row index of the top-1000 in rank order.
// ---------------------------------------------------------------------------
extern "C" __global__ __launch_bounds__(SORT_THREADS)
void topk_bitonic_kernel(const float* __restrict__ scores, int* __restrict__ top) {
  extern __shared__ unsigned long long key[];  // NPB * 8 = 131072 bytes
  const int b = blockIdx.x;
  const float* srow = scores + (size_t)b * NPB * NCLS;

#pragma unroll
  for (int s = 0; s < EPT; ++s) {
    unsigned i = threadIdx.x + s * SORT_THREADS;
    float fg = 1.0f - srow[(size_t)i * NCLS];  // background class = column 0
    unsigned bits = __float_as_uint(fg);       // fg in [0,1] -> monotone bits
    key[i] = ((unsigned long long)bits << 32) |
             (unsigned long long)(0xFFFFFFFFu - i);
  }
  __syncthreads();

  for (unsigned k = 2; k <= (unsigned)NPB; k <<= 1) {
    for (unsigned j = k >> 1; j > 0; j >>= 1) {
#pragma unroll
      for (int s = 0; s < EPT; ++s) {
        unsigned i = threadIdx.x + s * SORT_THREADS;
        unsigned p = i ^ j;
        if (p > i) {  // each disjoint pair handled exactly once
          unsigned long long a = key[i];
          unsigned long long c = key[p];
          bool up = ((i & k) == 0);            // descending where up==true
          if ((a < c) == up) { key[i] = c; key[p] = a; }
        }
      }
      __syncthreads();
    }
  }

  if (threadIdx.x < KTOP) {
    unsigned long long kk = key[threadIdx.x];
    unsigned local = 0xFFFFFFFFu - (unsigned)kk;   // recover candidate index
    top[b * KTOP + threadIdx.x] = b * NPB + (int)local;  // global row
  }
}

// ---------------------------------------------------------------------------
// Kernel 2: gather 64 kept rows per block through LDS using async b128 DMA.
// Scattered 16B chunks are pulled global->LDS (ASYNCcnt), then streamed out
// contiguously LDS->global. All addresses are 16B aligned (score row = 320B,
// box row = 16B, output offsets are multiples of 16B).
// ---------------------------------------------------------------------------
extern "C" __global__ __launch_bounds__(256)
void gather_async_kernel(const float* __restrict__ scores,
                         const float* __restrict__ boxes,
                         const int* __restrict__ top,
                         float* __restrict__ out) {
  __shared__ __align__(16) float sScores[ROWS_PER_BLOCK * NCLS];  // 20 KB
  __shared__ __align__(16) float sBoxes[ROWS_PER_BLOCK * 4];      // 1 KB
  __shared__ int sSrc[ROWS_PER_BLOCK];

  const int t = threadIdx.x;
  const int orow0 = blockIdx.x * ROWS_PER_BLOCK;  // first output row (rank)

  if (t < ROWS_PER_BLOCK) sSrc[t] = top[orow0 + t];
  __syncthreads();

  // Phase 1: async gather scattered rows into LDS.
  for (int c = t; c < CHUNKS_TOTAL; c += 256) {
    if (c < CHUNKS_SCORES) {
      int row = c / (NCLS / 4);
      int part = c % (NCLS / 4);
      const float* g = scores + (size_t)sSrc[row] * NCLS + part * 4;
      async_copy_g2l_b128(g, &sScores[c * 4]);   // c*4 == row*80 + part*4
    } else {
      int row = c - CHUNKS_SCORES;
      const float* g = boxes + (size_t)sSrc[row] * 4;
      async_copy_g2l_b128(g, &sBoxes[row * 4]);
    }
  }
  wait_async_zero();   // each wave waits for its own ASYNCcnt ...
  __syncthreads();     // ... then all waves' LDS data is visible

  float* outScores = out;                                 // [64000*80]
  float* outBI     = out + (size_t)BATCH * KTOP * NCLS;   // [64000]
  float* outBoxes  = outBI + (size_t)BATCH * KTOP;        // [64000*4]

  // Phase 2: async stream LDS out to contiguous output regions.
  for (int c = t; c < CHUNKS_TOTAL; c += 256) {
    if (c < CHUNKS_SCORES) {
      float* g = outScores + (size_t)orow0 * NCLS + c * 4;
      async_copy_l2g_b128(g, &sScores[c * 4]);
    } else {
      int row = c - CHUNKS_SCORES;
      float* g = outBoxes + (size_t)(orow0 + row) * 4;
      async_copy_l2g_b128(g, &sBoxes[row * 4]);
    }
  }

  // Batch indices (int values 0..63 are exact in f32).
  if (t < ROWS_PER_BLOCK) {
    int orow = orow0 + t;
    outBI[orow] = (float)(orow / KTOP);
  }
  wait_async_zero();  // drain async stores before wave retires
}

extern "C" void kernel_launch(void* const* d_in, const int* in_sizes, int n_in,
                              void* d_out, int out_size, void* d_ws, size_t ws_size,
                              hipStream_t stream) {
  (void)in_sizes; (void)n_in; (void)out_size; (void)ws_size;
  const float* scores = (const float*)d_in[0];
  // d_in[1] = batch_indices (repeat(arange(B), NPB)) — value == row / NPB,
  // reproduced analytically in the gather kernel.
  const float* boxes = (const float*)d_in[2];
  int* top = (int*)d_ws;  // 64000 ints = 256 KB of scratch

  // 128 KB dynamic LDS per workgroup (CDNA5 WGP has 320 KB).
  (void)hipFuncSetAttribute((const void*)topk_bitonic_kernel,
                            hipFuncAttributeMaxDynamicSharedMemorySize,
                            NPB * (int)sizeof(unsigned long long));

  topk_bitonic_kernel<<<BATCH, SORT_THREADS,
                        NPB * sizeof(unsigned long long), stream>>>(scores, top);

  gather_async_kernel<<<(BATCH * KTOP) / ROWS_PER_BLOCK, 256, 0, stream>>>(
      scores, boxes, top, (float*)d_out);
}